// Attention_19009525252630
// MI455X (gfx1250) — compile-verified
//
#include <hip/hip_runtime.h>
#include <hip/hip_bf16.h>
#include <cstdint>

// Problem constants (match reference)
#define HID   1024
#define HEADS 16
#define HDIM  64
#define SEQ   2048
#define BATCH 2
#define N3    3072   // 3*HID

typedef float v2f __attribute__((ext_vector_type(2)));
typedef float v8f __attribute__((ext_vector_type(8)));

// ---------------------------------------------------------------------------
// CDNA5 async global->LDS staging (ASYNCcnt path), with portable fallback.
// Probe result (round 2 diagnostics): the builtins take
//   b128: (AS1 int_vec16* src, AS3 dst, i32 offset, i32 cpol)
//   b32:  (AS1 int*       src, AS3 dst, i32 offset, i32 cpol)
// where AS1 prints as __device__ and AS3 as __shared__ in clang HIP mode.
// ---------------------------------------------------------------------------
#if __has_builtin(__builtin_amdgcn_global_load_async_to_lds_b128) && \
    __has_builtin(__builtin_amdgcn_global_load_async_to_lds_b32)  && \
    __has_builtin(__builtin_amdgcn_s_wait_asynccnt)
#define USE_ASYNC_LDS 1
#else
#define USE_ASYNC_LDS 0
#endif

typedef int gv4i __attribute__((vector_size(16)));
#define AS1P(T) T __attribute__((address_space(1))) *
#define AS3P(T) T __attribute__((address_space(3))) *

// On amdgcn the global aperture is identity (flat addr == AS1 addr) and the
// low 32 bits of a flat LDS address are the AS3 offset, so integer round-trips
// are architecturally valid ways to produce the address-space pointers.
__device__ __forceinline__ void async_copy16(const float* g, float* l) {
#if USE_ASYNC_LDS
  __builtin_amdgcn_global_load_async_to_lds_b128(
      (AS1P(gv4i))(uintptr_t)g, (AS3P(gv4i))(uintptr_t)l, 0, 0);
#else
  *(float4*)l = *(const float4*)g;
#endif
}
__device__ __forceinline__ void async_copy4(const float* g, float* l) {
#if USE_ASYNC_LDS
  __builtin_amdgcn_global_load_async_to_lds_b32(
      (AS1P(int))(uintptr_t)g, (AS3P(int))(uintptr_t)l, 0, 0);
#else
  *l = *g;
#endif
}
__device__ __forceinline__ void wait_async0() {
#if USE_ASYNC_LDS
  __builtin_amdgcn_s_wait_asynccnt(0);
#endif
}

// D(16x16,f32) = A(16x4,f32) * B(4x16,f32) + C
// fp32 A layout (ISA 7.12.2): lanes 0-15 hold K=0 (v0), K=1 (v1) for M=lane;
// lanes 16-31 hold K=2,K=3.  B mirrors with N=lane&15.
// C/D: vgpr r holds row (r + 8*(lane>>4)), col (lane&15).
static __device__ __forceinline__ v8f wmma4(v2f a, v2f b, v8f c) {
  return __builtin_amdgcn_wmma_f32_16x16x4_f32(
      false, a, false, b, (short)0, c, false, false);
}

// ---------------------------------------------------------------------------
// GEMM:  out[M=4096, N] = A[4096, 1024] @ W[1024, N] + bias[N]
// Block: 8 waves; block tile 128(M) x 64(N); wave tile 16(M) x 64(N).
// K staged in double-buffered LDS chunks of 32 via async loads:
//   sA [128][32] row-major (b128 async, coalesced),
//   sB [64][32]  (W chunk transposed via per-lane b32 async) so B fragments
//   are contiguous ds_load_b64 reads.
// ---------------------------------------------------------------------------
template <int N>
__global__ __launch_bounds__(256) void gemm_bias(const float* __restrict__ A,
                                                 const float* __restrict__ W,
                                                 const float* __restrict__ bias,
                                                 float* __restrict__ out) {
  constexpr int KC = 32;
  __shared__ float sA[2][128 * KC];   // 16 KB each
  __shared__ float sB[2][64 * KC];    //  8 KB each (transposed: [n][k])
  const int t    = threadIdx.x;
  const int wv   = t >> 5;
  const int lane = t & 31;
  const int half = lane >> 4;
  const int ln   = lane & 15;
  const int tm   = blockIdx.x / (N / 64);
  const int tn   = blockIdx.x % (N / 64);

  const float* Ablk = A + (size_t)tm * 128 * HID;
  const float* Wblk = W + tn * 64;

  auto stage = [&](int kc, int buf) {
    const int k0 = kc * KC;
    // A chunk: 128x32 floats = 1024 b128 transfers, 4 per thread
#pragma unroll
    for (int i = 0; i < 4; ++i) {
      const int id  = t + 256 * i;       // 0..1023
      const int row = id >> 3;
      const int col = (id & 7) * 4;
      async_copy16(Ablk + (size_t)row * HID + k0 + col,
                   &sA[buf][row * KC + col]);
    }
    // W chunk transposed: 32x64 floats = 2048 b32 transfers, 8 per thread
#pragma unroll
    for (int i = 0; i < 8; ++i) {
      const int id = t + 256 * i;        // 0..2047
      const int kr = id >> 6;            // 0..31
      const int n  = id & 63;
      async_copy4(Wblk + (size_t)(k0 + kr) * N + n,
                  &sB[buf][n * KC + kr]);
    }
  };

  v8f acc[4];
#pragma unroll
  for (int c = 0; c < 4; ++c) acc[c] = (v8f){};

  stage(0, 0);
  for (int kc = 0; kc < HID / KC; ++kc) {
    const int buf = kc & 1;
    wait_async0();
    __syncthreads();                     // tile ready + prev reads of buf^1 done
    if (kc + 1 < HID / KC) stage(kc + 1, buf ^ 1);

    const float* a_s = &sA[buf][(wv * 16 + ln) * KC];
#pragma unroll
    for (int ks = 0; ks < KC; ks += 4) {
      const int ka = ks + 2 * half;
      v2f a = *(const v2f*)(a_s + ka);
#pragma unroll
      for (int c = 0; c < 4; ++c) {
        v2f b = *(const v2f*)(&sB[buf][(c * 16 + ln) * KC + ka]);
        acc[c] = wmma4(a, b, acc[c]);
      }
    }
  }

  const int row0 = tm * 128 + wv * 16 + 8 * half;
#pragma unroll
  for (int c = 0; c < 4; ++c) {
    const int col = tn * 64 + c * 16 + ln;
    const float bv = bias[col];
    float* orow = out + (size_t)row0 * N + col;
#pragma unroll
    for (int r = 0; r < 8; ++r)
      orow[(size_t)r * N] = acc[c][r] + bv;
  }
}

// ---------------------------------------------------------------------------
// In-place RoPE on q,k slices. Reference indexes the rope table by HEAD
// (cos_tab[:HEADS]) -> angle = h * inv_freq[j], independent of position.
// ---------------------------------------------------------------------------
__global__ __launch_bounds__(256) void rope_kernel(float* __restrict__ qkv) {
  const int idx = blockIdx.x * blockDim.x + threadIdx.x;  // B*S*H*32
  const int j = idx & 31;
  const int h = (idx >> 5) & (HEADS - 1);
  const int t = idx >> 9;
  if (t >= BATCH * SEQ) return;
  const float invf = __expf(-0.2878232398796653f * (float)j); // 10000^(-2j/64)
  float sn, cs;
  __sincosf((float)h * invf, &sn, &cs);
  const size_t base = (size_t)t * N3 + h * 192;
  float q1 = qkv[base + j], q2 = qkv[base + j + 32];
  qkv[base + j]      = q1 * cs - q2 * sn;
  qkv[base + j + 32] = q2 * cs + q1 * sn;
  float k1 = qkv[base + 64 + j], k2 = qkv[base + 96 + j];
  qkv[base + 64 + j] = k1 * cs - k2 * sn;
  qkv[base + 96 + j] = k2 * cs + k1 * sn;
}

// ---------------------------------------------------------------------------
// Flash attention. grid = (SEQ/128, HEADS, BATCH), 8 waves/block.
// Wave w owns q rows [q0, q0+16); all waves share (b,h), so K/V tiles are
// cooperatively async-staged once per block into double-buffered LDS and the
// next tile's load is issued before computing the current one.
// ---------------------------------------------------------------------------
__global__ __launch_bounds__(256) void flash_attn(const float* __restrict__ qkv,
                                                  float* __restrict__ attn_out) {
  __shared__ float kv[2][32 * 64];      // rows 0-15: K tile, 16-31: V tile
  __shared__ float plds[8][16 * 18];    // per-wave P transpose slab
  const int t    = threadIdx.x;
  const int wv   = t >> 5;
  const int lane = t & 31;
  const int half = lane >> 4;
  const int ln   = lane & 15;
  const int b  = blockIdx.z;
  const int h  = blockIdx.y;
  const int q0 = (blockIdx.x * 8 + wv) * 16;
  const float* hb = qkv + (size_t)(b * SEQ) * N3 + h * 192;   // head base

  auto stage = [&](int kt, int buf) {
    // 16 K rows + 16 V rows, 64 floats each = 512 b128 transfers, 2/thread
#pragma unroll
    for (int i = 0; i < 2; ++i) {
      const int id  = t + 256 * i;      // 0..511
      const int row = id >> 4;          // 0..31
      const int off = (id & 15) * 4;
      const float* g = hb + (size_t)(kt * 16 + (row & 15)) * N3 +
                       ((row < 16) ? 64 : 128) + off;
      async_copy16(g, &kv[buf][row * 64 + off]);
    }
  };

  // Preload this wave's Q tile as 16 A-fragments (all 64 head dims).
  v2f qa[16];
  {
    const float* qrow = hb + (size_t)(q0 + ln) * N3;
#pragma unroll
    for (int kk = 0; kk < 16; ++kk)
      qa[kk] = *(const v2f*)(qrow + kk * 4 + 2 * half);
  }

  float mrow[8], lrow[8];
  v8f o[4];
#pragma unroll
  for (int r = 0; r < 8; ++r) { mrow[r] = -3.0e38f; lrow[r] = 0.0f; }
#pragma unroll
  for (int c = 0; c < 4; ++c) o[c] = (v8f){};
  float* myp = &plds[wv][0];

  stage(0, 0);
  for (int kt = 0; kt < SEQ / 16; ++kt) {
    const int buf = kt & 1;
    wait_async0();
    __syncthreads();                    // tile ready; prior reads of buf^1 done
    if (kt + 1 < SEQ / 16) stage(kt + 1, buf ^ 1);

    // ---- S = Q @ K^T, K fragments from LDS (contiguous ds_load_b64) ----
    v8f sacc = {};
#pragma unroll
    for (int kk = 0; kk < 16; ++kk) {
      v2f kb = *(const v2f*)(&kv[buf][ln * 64 + kk * 4 + 2 * half]);
      sacc = wmma4(qa[kk], kb, sacc);
    }

    // ---- online softmax: row = r + 8*half, 16 cols spread over 16 lanes ----
#pragma unroll
    for (int r = 0; r < 8; ++r) {
      float s = sacc[r] * 0.125f;       // 1/sqrt(64)
      float rmax = s;
      rmax = fmaxf(rmax, __shfl_xor(rmax, 1, 32));
      rmax = fmaxf(rmax, __shfl_xor(rmax, 2, 32));
      rmax = fmaxf(rmax, __shfl_xor(rmax, 4, 32));
      rmax = fmaxf(rmax, __shfl_xor(rmax, 8, 32));
      const float mnew = fmaxf(mrow[r], rmax);
      const float p = __expf(s - mnew);
      float rsum = p;
      rsum += __shfl_xor(rsum, 1, 32);
      rsum += __shfl_xor(rsum, 2, 32);
      rsum += __shfl_xor(rsum, 4, 32);
      rsum += __shfl_xor(rsum, 8, 32);
      const float alpha = __expf(mrow[r] - mnew);
      lrow[r] = lrow[r] * alpha + rsum;
      mrow[r] = mnew;
#pragma unroll
      for (int c = 0; c < 4; ++c) o[c][r] *= alpha;
      myp[(r + 8 * half) * 18 + ln] = p;   // D-layout -> LDS
    }
    asm volatile("s_wait_dscnt 0" ::: "memory");  // cross-lane RAW in-wave

    // ---- O += P @ V, P reread in A layout, V fragments from LDS ----
#pragma unroll
    for (int c = 0; c < 4; ++c) {
#pragma unroll
      for (int kk = 0; kk < 4; ++kk) {
        const int kd = kk * 4 + 2 * half;
        v2f pa = *(const v2f*)(myp + ln * 18 + kd);
        v2f vb;
        vb.x = kv[buf][(16 + kd) * 64 + c * 16 + ln];
        vb.y = kv[buf][(16 + kd + 1) * 64 + c * 16 + ln];
        o[c] = wmma4(pa, vb, o[c]);
      }
    }
  }

  // ---- normalize, write attn output as [B*S, HID] row-major ----
  float* orow = attn_out + (size_t)(b * SEQ + q0 + 8 * half) * HID + h * HDIM;
#pragma unroll
  for (int c = 0; c < 4; ++c)
#pragma unroll
    for (int r = 0; r < 8; ++r)
      orow[(size_t)r * HID + c * 16 + ln] = o[c][r] / lrow[r];
}

// ---------------------------------------------------------------------------
extern "C" void kernel_launch(void* const* d_in, const int* in_sizes, int n_in,
                              void* d_out, int out_size, void* d_ws, size_t ws_size,
                              hipStream_t stream) {
  (void)in_sizes; (void)n_in; (void)out_size; (void)ws_size;
  const float* x     = (const float*)d_in[0];
  const float* qkv_w = (const float*)d_in[1];
  const float* qkv_b = (const float*)d_in[2];
  const float* out_w = (const float*)d_in[3];
  const float* out_b = (const float*)d_in[4];
  float* out = (float*)d_out;

  // Workspace: qkv [4096,3072] f32 (48 MB) then attn [4096,1024] f32 (16 MB)
  float* qkv  = (float*)d_ws;
  float* attn = qkv + (size_t)BATCH * SEQ * N3;

  // 1) QKV projection + bias: block tile 128x64 -> 32*48 blocks
  gemm_bias<N3><<<dim3((BATCH * SEQ / 128) * (N3 / 64)), 256, 0, stream>>>(
      x, qkv_w, qkv_b, qkv);

  // 2) head-indexed RoPE on q,k (in place)
  rope_kernel<<<(BATCH * SEQ * HEADS * 32) / 256, 256, 0, stream>>>(qkv);

  // 3) flash attention -> [B*S, HID]
  flash_attn<<<dim3(SEQ / 128, HEADS, BATCH), 256, 0, stream>>>(qkv, attn);

  // 4) output projection: 32*16 blocks
  gemm_bias<HID><<<dim3((BATCH * SEQ / 128) * (HID / 64)), 256, 0, stream>>>(
      attn, out_w, out_b, out);
}